// EssenceNetClassifier_60748017434852
// MI455X (gfx1250) — compile-verified
//
#include <hip/hip_runtime.h>
#include <math.h>

typedef float v2f __attribute__((ext_vector_type(2)));
typedef float v8f __attribute__((ext_vector_type(8)));

// ---------------- model constants ----------------
static constexpr int FCH[8]   = {1027, 515, 259, 131, 67, 35, 19, 11};
static constexpr int ACCUMC[8]= {1027, 1542, 1801, 1932, 1999, 2034, 2053, 2064};
static constexpr int OUTSC[8] = {1024, 512, 256, 128, 64, 32, 16, 8};
static constexpr int KSV[8]   = {256, 128, 64, 32, 16, 8, 4, 3};
static constexpr int STV[8]   = {256, 128, 64, 32, 16, 8, 4, 2};
static constexpr int PDV[8]   = {0, 0, 0, 0, 0, 0, 0, 1};

#define FOFF_INIT {0LL, 2054LL, 6174LL, 14462LL, 31230LL, 65534LL, 137214LL, 292862LL}
static constexpr long long H_FOFF[8] = FOFF_INIT;          // host copy
__constant__ long long D_FOFF[8] = FOFF_INIT;              // device copy
__constant__ int       D_FCH[8]  = {1027, 515, 259, 131, 67, 35, 19, 11};

// ---------------- workspace layout (float offsets) ----------------
static constexpr long long GRAY_O   = 0;                                // 2*65536
static constexpr long long POOL_O   = GRAY_O  + 131072;                 // 131070 (pad 131072)
static constexpr long long FEATS_O  = POOL_O  + 131072;                 // 653310 (pad 653376)
static constexpr long long BBACC_O  = FEATS_O + 653376;                 // split-K backbone partials
static constexpr long long Y1_O     = BBACC_O + 524288;                 // 1032*4096
static constexpr long long Y2_O     = Y1_O    + 1032LL * 4096;          // 516*4096
static constexpr long long ALLF_O   = Y2_O    + 516LL * 4096;           // 2064*4096 gathered chunk
static constexpr long long PREVA_O  = ALLF_O  + 2064LL * 4096;          // 2*1000*4096
static constexpr long long PREVB_O  = PREVA_O + 2LL * 1000 * 4096;
static constexpr long long FINAL_O  = PREVB_O + 2LL * 1000 * 4096;      // 2*1000*16384
static constexpr long long STATS_O  = FINAL_O + 2LL * 1000 * 16384;     // 8 doubles (even)
static constexpr long long PART_O   = STATS_O + 16;                     // 512 doubles (even)
static constexpr long long PMAX_O   = PART_O  + 1024;
static constexpr long long ARGC_O   = PMAX_O  + 16384;
static constexpr long long SCORES_O = ARGC_O  + 16384;
static constexpr long long KEYS_O   = SCORES_O + 16384;
static constexpr long long MISC_O   = KEYS_O  + 16384;
static constexpr int CHUNK = 4096;

static constexpr long long poolOffL(int L) {
  long long o = 0;
  for (int M = 7; M > L; --M) o += 6LL << (2 * M);
  return o;
}

// ---------------- small helpers ----------------
__device__ __forceinline__ float bn_silu_f(float acc, float g, float b) {
  float v = acc * 0.9999950000374997f * g + b;   // 1/sqrt(1+1e-5)
  return v / (1.0f + __expf(-v));
}

__device__ __forceinline__ v8f wmma4(v2f a, v2f b, v8f c) {
  return __builtin_amdgcn_wmma_f32_16x16x4_f32(false, a, false, b, (short)0, c, false, false);
}

__device__ __forceinline__ int imin_(int a, int b) { return a < b ? a : b; }
__device__ __forceinline__ float zsel_(bool ok, float v) { return ok ? v : 0.f; }

// ---------------- elementwise / pooling kernels ----------------
__global__ void k_gray(const float* __restrict__ x, float* __restrict__ gray) {
  int tid = blockIdx.x * blockDim.x + threadIdx.x;
  if (tid >= 2 * 65536) return;
  int p = tid & 65535, b = tid >> 16;
  const float* xb = x + (long long)b * 3 * 65536;
  gray[tid] = 0.299f * xb[p] + 0.587f * xb[65536 + p] + 0.114f * xb[2 * 65536 + p];
}

__global__ void k_pool_half(const float* __restrict__ in, float* __restrict__ out, int S) {
  // in: [6, 2S, 2S] -> out: [6, S, S] (2x2 average)
  int tid = blockIdx.x * blockDim.x + threadIdx.x;
  int total = 6 * S * S;
  if (tid >= total) return;
  int x = tid % S; int t = tid / S; int y = t % S; int c = t / S;
  const float* ip = in + ((long long)c * 2 * S + 2 * y) * 2 * S + 2 * x;
  out[tid] = 0.25f * (ip[0] + ip[1] + ip[2 * S] + ip[2 * S + 1]);
}

__global__ void k_copy_rc(const float* __restrict__ pool, float* __restrict__ feat, int H, int FC) {
  int tid = blockIdx.x * blockDim.x + threadIdx.x;
  int total = 6 * H * H;
  if (tid >= total) return;
  int x = tid % H; int t = tid / H; int y = t % H; t /= H;
  int c = t % 3; int b = t / 3;
  feat[(((long long)b * FC + c) * H + y) * H + x] = pool[tid];
}

// ---------------- backbone convs ----------------
// direct version (used for i>=4, many output pixels)
__global__ void k_backbone(const float* __restrict__ gray, const float* __restrict__ w,
                           const float* __restrict__ g, const float* __restrict__ bb_,
                           float* __restrict__ feat, int OC, int Ks, int St, int Pd,
                           int Ho, int FC) {
  long long tid = (long long)blockIdx.x * blockDim.x + threadIdx.x;
  long long total = 2LL * OC * Ho * Ho;
  if (tid >= total) return;
  int x = (int)(tid % Ho); long long t = tid / Ho;
  int y = (int)(t % Ho); t /= Ho;
  int oc = (int)(t % OC); int b = (int)(t / OC);
  const float* gb = gray + (long long)b * 65536;
  const float* wr = w + (long long)oc * Ks * Ks;
  float acc = 0.f;
  int iy0 = y * St - Pd, ix0 = x * St - Pd;
  for (int ky = 0; ky < Ks; ++ky) {
    int iy = iy0 + ky; if (iy < 0 || iy >= 256) continue;
    for (int kx = 0; kx < Ks; ++kx) {
      int ix = ix0 + kx; if (ix < 0 || ix >= 256) continue;
      acc += gb[iy * 256 + ix] * wr[ky * Ks + kx];
    }
  }
  feat[(((long long)b * FC + 3 + oc) * Ho + y) * Ho + x] = bn_silu_f(acc, g[oc], bb_[oc]);
}

// split-K version (i<=3: no padding, stride==Ks): one thread per (output, ky) row
__global__ void k_backbone_split(const float* __restrict__ gray, const float* __restrict__ w,
                                 float* __restrict__ part, int OC, int Ks, int St, int Ho) {
  long long tid = (long long)blockIdx.x * blockDim.x + threadIdx.x;
  long long total = 2LL * OC * Ho * Ho * Ks;
  if (tid >= total) return;
  int ky = (int)(tid % Ks);
  long long o = tid / Ks;
  int x = (int)(o % Ho); long long t = o / Ho;
  int y = (int)(t % Ho); t /= Ho;
  int oc = (int)(t % OC); int b = (int)(t / OC);
  const float* gr = gray + (long long)b * 65536 + (long long)(y * St + ky) * 256 + x * St;
  const float* wr = w + ((long long)oc * Ks + ky) * Ks;
  float s = 0.f;
  for (int kx = 0; kx < Ks; ++kx) s += gr[kx] * wr[kx];
  part[tid] = s;
}

__global__ void k_backbone_fin(const float* __restrict__ part, const float* __restrict__ g,
                               const float* __restrict__ bb_, float* __restrict__ feat,
                               int OC, int Ks, int Ho, int FC) {
  long long o = (long long)blockIdx.x * blockDim.x + threadIdx.x;
  long long total = 2LL * OC * Ho * Ho;
  if (o >= total) return;
  int x = (int)(o % Ho); long long t = o / Ho;
  int y = (int)(t % Ho); t /= Ho;
  int oc = (int)(t % OC); int b = (int)(t / OC);
  const float* pr = part + o * Ks;
  float acc = 0.f;
  for (int ky = 0; ky < Ks; ++ky) acc += pr[ky];   // fixed order -> deterministic
  feat[(((long long)b * FC + 3 + oc) * Ho + y) * Ho + x] = bn_silu_f(acc, g[oc], bb_[oc]);
}

// ---------------- gather the virtual concat(up(feats[0..i-1]), feats[i]) chunk ----------------
__global__ void k_gather(const float* __restrict__ fb, float* __restrict__ dst,
                         int K, int headi, int b, int chunkBase, int CP) {
  int tid = blockIdx.x * blockDim.x + threadIdx.x;
  long long total = (long long)K * CP;
  if (tid >= total) return;
  int p = tid % CP;
  int ci = tid / CP;
  int gp = chunkBase + p;
  int Hi = 1 << headi;
  int y = gp >> headi, x = gp & (Hi - 1);
  int j = 0, off = ci;
  while (off >= D_FCH[j]) { off -= D_FCH[j]; ++j; }
  int sh = headi - j;
  int Hj = 1 << j;
  dst[tid] = fb[D_FOFF[j] + (((long long)b * D_FCH[j] + off) * Hj + (y >> sh)) * Hj + (x >> sh)];
}

// ---------------- WMMA GEMM (fp32 16x16x4), 32x32 tile per wave ----------------
// Out[m, p] = sum_k W[m,k] * Bm[k, p]; mode 0 = BN+SiLU, mode 1 = +bias
__global__ void __launch_bounds__(32) k_gemm(
    const float* __restrict__ W, const float* __restrict__ Bm, float* __restrict__ Out,
    const float* __restrict__ gma, const float* __restrict__ bta,
    int OC, int K, int CP, int ldb, int ldo, int mode) {
  int lane = threadIdx.x;
  int half = lane >> 4, sub = lane & 15;
  int oc0 = blockIdx.x * 32, p0 = blockIdx.y * 32;
  v8f z = {0, 0, 0, 0, 0, 0, 0, 0};
  v8f acc[2][2] = {{z, z}, {z, z}};
  int K4 = K & ~3;
  bool full = (oc0 + 32 <= OC) && (p0 + 32 <= CP);

  if (full) {
    const float* wr0 = W + (long long)(oc0 + sub) * K;
    const float* wr1 = W + (long long)(oc0 + 16 + sub) * K;
    const float* bc0 = Bm + p0 + sub;
    const float* bc1 = bc0 + 16;
#pragma unroll 2
    for (int k0 = 0; k0 < K4; k0 += 4) {
      int ka = k0 + 2 * half;
      long long r0 = (long long)ka * ldb, r1 = r0 + ldb;
      v2f a0, a1, b0, b1;
      a0.x = wr0[ka]; a0.y = wr0[ka + 1];
      a1.x = wr1[ka]; a1.y = wr1[ka + 1];
      b0.x = bc0[r0]; b0.y = bc0[r1];
      b1.x = bc1[r0]; b1.y = bc1[r1];
      acc[0][0] = wmma4(a0, b0, acc[0][0]);
      acc[0][1] = wmma4(a0, b1, acc[0][1]);
      acc[1][0] = wmma4(a1, b0, acc[1][0]);
      acc[1][1] = wmma4(a1, b1, acc[1][1]);
    }
    if (K4 < K) {  // K tail: unconditional clamped loads, then zero-select
      int ka = K4 + 2 * half;
      int kc0 = imin_(ka, K - 1), kc1 = imin_(ka + 1, K - 1);
      bool k0ok = ka < K, k1ok = ka + 1 < K;
      long long r0 = (long long)kc0 * ldb, r1 = (long long)kc1 * ldb;
      float la0x = wr0[kc0], la0y = wr0[kc1];
      float la1x = wr1[kc0], la1y = wr1[kc1];
      float lb0x = bc0[r0],  lb0y = bc0[r1];
      float lb1x = bc1[r0],  lb1y = bc1[r1];
      v2f a0, a1, b0, b1;
      a0.x = zsel_(k0ok, la0x); a0.y = zsel_(k1ok, la0y);
      a1.x = zsel_(k0ok, la1x); a1.y = zsel_(k1ok, la1y);
      b0.x = zsel_(k0ok, lb0x); b0.y = zsel_(k1ok, lb0y);
      b1.x = zsel_(k0ok, lb1x); b1.y = zsel_(k1ok, lb1y);
      acc[0][0] = wmma4(a0, b0, acc[0][0]);
      acc[0][1] = wmma4(a0, b1, acc[0][1]);
      acc[1][0] = wmma4(a1, b0, acc[1][0]);
      acc[1][1] = wmma4(a1, b1, acc[1][1]);
    }
  } else {  // edge tiles: clamped addresses, unconditional loads, cndmask zero-select
    int mA0 = oc0 + sub, mA1 = mA0 + 16;
    bool m0ok = mA0 < OC, m1ok = mA1 < OC;
    int nB0 = p0 + sub, nB1 = nB0 + 16;
    bool n0ok = nB0 < CP, n1ok = nB1 < CP;
    const float* wr0 = W + (long long)imin_(mA0, OC - 1) * K;
    const float* wr1 = W + (long long)imin_(mA1, OC - 1) * K;
    const float* bc0 = Bm + imin_(nB0, CP - 1);
    const float* bc1 = Bm + imin_(nB1, CP - 1);
    for (int k0 = 0; k0 < K; k0 += 4) {
      int ka = k0 + 2 * half;
      int kc0 = imin_(ka, K - 1), kc1 = imin_(ka + 1, K - 1);
      bool k0ok = ka < K, k1ok = ka + 1 < K;
      long long r0 = (long long)kc0 * ldb, r1 = (long long)kc1 * ldb;
      // all addresses in-bounds: issue loads unconditionally (no EXEC juggling)
      float la0x = wr0[kc0], la0y = wr0[kc1];
      float la1x = wr1[kc0], la1y = wr1[kc1];
      float lb0x = bc0[r0],  lb0y = bc0[r1];
      float lb1x = bc1[r0],  lb1y = bc1[r1];
      v2f a0, a1, b0, b1;
      a0.x = zsel_(m0ok && k0ok, la0x); a0.y = zsel_(m0ok && k1ok, la0y);
      a1.x = zsel_(m1ok && k0ok, la1x); a1.y = zsel_(m1ok && k1ok, la1y);
      b0.x = zsel_(n0ok && k0ok, lb0x); b0.y = zsel_(n0ok && k1ok, lb0y);
      b1.x = zsel_(n1ok && k0ok, lb1x); b1.y = zsel_(n1ok && k1ok, lb1y);
      acc[0][0] = wmma4(a0, b0, acc[0][0]);
      acc[0][1] = wmma4(a0, b1, acc[0][1]);
      acc[1][0] = wmma4(a1, b0, acc[1][0]);
      acc[1][1] = wmma4(a1, b1, acc[1][1]);
    }
  }

#pragma unroll
  for (int r = 0; r < 2; ++r) {
#pragma unroll
    for (int c = 0; c < 2; ++c) {
#pragma unroll
      for (int v = 0; v < 8; ++v) {
        int m = oc0 + 16 * r + v + half * 8;
        int p = p0 + 16 * c + sub;
        if (m < OC && p < CP) {
          float val = acc[r][c][v];
          if (mode == 0) val = bn_silu_f(val, gma[m], bta[m]);
          else           val = val + bta[m];
          Out[(long long)m * ldo + p] = val;
        }
      }
    }
  }
}

// ---------------- normalization stats (deterministic two-stage) ----------------
__global__ void k_stats(const float* __restrict__ src, long long N, double mult,
                        double* __restrict__ part) {
  __shared__ double s1[256], s2[256];
  long long tid = (long long)blockIdx.x * blockDim.x + threadIdx.x;
  long long stride = (long long)gridDim.x * blockDim.x;
  double a = 0, b = 0;
  for (long long i = tid; i < N; i += stride) { double v = src[i]; a += v; b += v * v; }
  s1[threadIdx.x] = a; s2[threadIdx.x] = b;
  __syncthreads();
  for (int o = 128; o > 0; o >>= 1) {
    if ((int)threadIdx.x < o) { s1[threadIdx.x] += s1[threadIdx.x + o]; s2[threadIdx.x] += s2[threadIdx.x + o]; }
    __syncthreads();
  }
  if (threadIdx.x == 0) { part[2 * blockIdx.x] = mult * s1[0]; part[2 * blockIdx.x + 1] = mult * s2[0]; }
}

__global__ void __launch_bounds__(256) k_stats_fin(const double* __restrict__ part, int nblk,
                                                   double* __restrict__ outp) {
  __shared__ double s1[256], s2[256];
  double a = 0, b = 0;
  for (int i = threadIdx.x; i < nblk; i += 256) { a += part[2 * i]; b += part[2 * i + 1]; }
  s1[threadIdx.x] = a; s2[threadIdx.x] = b;
  __syncthreads();
  for (int o = 128; o > 0; o >>= 1) {
    if ((int)threadIdx.x < o) { s1[threadIdx.x] += s1[threadIdx.x + o]; s2[threadIdx.x] += s2[threadIdx.x + o]; }
    __syncthreads();
  }
  if (threadIdx.x == 0) { outp[0] = s1[0]; outp[1] = s2[0]; }
}

__global__ void k_combine(const float* __restrict__ prev, const float* __restrict__ li,
                          float* __restrict__ cur, const double* __restrict__ st,
                          int H, double Nn) {
  long long tid = (long long)blockIdx.x * blockDim.x + threadIdx.x;
  long long per = 1000LL * H * H;
  if (tid >= 2 * per) return;
  int b = (int)(tid / per);
  long long r = tid - (long long)b * per;
  int c = (int)(r / (H * H));
  int pix = (int)(r % (H * H));
  int y = pix / H, x = pix % H;
  int Hp = H >> 1;
  const double* sb = st + b * 4;
  double mP = sb[0] / Nn, mL = sb[2] / Nn;
  float dP = (float)sqrt((sb[1] - sb[0] * mP) / (Nn - 1.0)) + 1e-5f;
  float dL = (float)sqrt((sb[3] - sb[2] * mL) / (Nn - 1.0)) + 1e-5f;
  float pv = prev[((long long)b * 1000 + c) * Hp * Hp + (long long)(y >> 1) * Hp + (x >> 1)];
  float lv = li[tid];
  cur[tid] = (pv - (float)mP) / dP + (lv - (float)mL) / dL;
}

// ---------------- selection (_select_one) ----------------
__global__ void k_pixmax(const float* __restrict__ lb, float* __restrict__ pmax,
                         int* __restrict__ argc, int P) {
  int p = blockIdx.x * blockDim.x + threadIdx.x;
  if (p >= P) return;
  float best = lb[p]; int bi = 0;
  for (int c = 1; c < 1000; ++c) {
    float v = lb[(long long)c * P + p];
    if (v > best) { best = v; bi = c; }
  }
  pmax[p] = best; argc[p] = bi;
}

__global__ void __launch_bounds__(1024) k_findpos(const float* __restrict__ pmax,
                                                  const int* __restrict__ argc,
                                                  int* __restrict__ miscI, int P) {
  __shared__ float sv[1024];
  __shared__ int si[1024];
  float best = -__builtin_inff(); int bi = 0x7fffffff;
  for (int p = threadIdx.x; p < P; p += 1024) {
    float v = pmax[p];
    if (v > best || (v == best && p < bi)) { best = v; bi = p; }
  }
  sv[threadIdx.x] = best; si[threadIdx.x] = bi;
  __syncthreads();
  for (int o = 512; o > 0; o >>= 1) {
    if ((int)threadIdx.x < o) {
      if (sv[threadIdx.x + o] > sv[threadIdx.x] ||
          (sv[threadIdx.x + o] == sv[threadIdx.x] && si[threadIdx.x + o] < si[threadIdx.x])) {
        sv[threadIdx.x] = sv[threadIdx.x + o]; si[threadIdx.x] = si[threadIdx.x + o];
      }
    }
    __syncthreads();
  }
  if (threadIdx.x == 0) miscI[0] = argc[si[0]];
}

__global__ void k_scores(const float* __restrict__ lb, const int* __restrict__ argc,
                         const int* __restrict__ miscI, float* __restrict__ scores,
                         unsigned* __restrict__ keys, int P) {
  int p = blockIdx.x * blockDim.x + threadIdx.x;
  if (p >= P) return;
  int cid = miscI[0];
  float s = lb[(long long)cid * P + p];
  scores[p] = s;
  unsigned u = __float_as_uint(s);
  unsigned key = (u & 0x80000000u) ? ~u : (u | 0x80000000u);
  keys[p] = (argc[p] == cid) ? key : 0xFFFFFFFFu;
}

__global__ void __launch_bounds__(1024) k_median(const unsigned* __restrict__ keys,
                                                  const int* __restrict__ argc,
                                                  const int* __restrict__ miscI,
                                                  float* __restrict__ miscF, int P) {
  __shared__ int cnt[1024];
  __shared__ unsigned sprefix;
  __shared__ int sr;
  int cid = miscI[0];
  int local = 0;
  for (int p = threadIdx.x; p < P; p += 1024) if (argc[p] == cid) local++;
  cnt[threadIdx.x] = local;
  __syncthreads();
  for (int o = 512; o > 0; o >>= 1) {
    if ((int)threadIdx.x < o) cnt[threadIdx.x] += cnt[threadIdx.x + o];
    __syncthreads();
  }
  if (threadIdx.x == 0) { sr = (cnt[0] - 1) >> 1; sprefix = 0u; }
  __syncthreads();
  for (int bit = 31; bit >= 0; --bit) {
    unsigned pref = sprefix;
    int c0 = 0;
    for (int p = threadIdx.x; p < P; p += 1024) {
      unsigned kk = keys[p];
      bool upper = (bit == 31) ? true : ((kk >> (bit + 1)) == (pref >> (bit + 1)));
      if (upper && ((kk >> bit) & 1u) == 0u) c0++;
    }
    cnt[threadIdx.x] = c0;
    __syncthreads();
    for (int o = 512; o > 0; o >>= 1) {
      if ((int)threadIdx.x < o) cnt[threadIdx.x] += cnt[threadIdx.x + o];
      __syncthreads();
    }
    if (threadIdx.x == 0) {
      if (sr >= cnt[0]) { sr -= cnt[0]; sprefix |= (1u << bit); }
    }
    __syncthreads();
  }
  if (threadIdx.x == 0) {
    unsigned key = sprefix;
    unsigned u = (key & 0x80000000u) ? (key ^ 0x80000000u) : ~key;
    miscF[0] = __uint_as_float(u);
  }
}

__global__ void __launch_bounds__(1024) k_argmin(const float* __restrict__ scores,
                                                  const int* __restrict__ argc,
                                                  int* __restrict__ miscI,
                                                  const float* __restrict__ miscF, int P) {
  __shared__ float sv[1024];
  __shared__ int si[1024];
  float med = miscF[0];
  int cid = miscI[0];
  float bd = __builtin_inff(); int bi = 0x7fffffff;
  for (int p = threadIdx.x; p < P; p += 1024) {
    if (argc[p] == cid) {
      float d = fabsf(scores[p] - med);
      if (d < bd || (d == bd && p < bi)) { bd = d; bi = p; }
    }
  }
  sv[threadIdx.x] = bd; si[threadIdx.x] = bi;
  __syncthreads();
  for (int o = 512; o > 0; o >>= 1) {
    if ((int)threadIdx.x < o) {
      if (sv[threadIdx.x + o] < sv[threadIdx.x] ||
          (sv[threadIdx.x + o] == sv[threadIdx.x] && si[threadIdx.x + o] < si[threadIdx.x])) {
        sv[threadIdx.x] = sv[threadIdx.x + o]; si[threadIdx.x] = si[threadIdx.x + o];
      }
    }
    __syncthreads();
  }
  if (threadIdx.x == 0) miscI[1] = si[0];
}

__global__ void k_writeout(const float* __restrict__ lb, const int* __restrict__ miscI,
                           float* __restrict__ out, int P) {
  int c = blockIdx.x * blockDim.x + threadIdx.x;
  if (c < 1000) out[c] = lb[(long long)c * P + miscI[1]];
}

// ---------------- host orchestration ----------------
static inline int cdiv(long long a, long long b) { return (int)((a + b - 1) / b); }

extern "C" void kernel_launch(void* const* d_in, const int* in_sizes, int n_in,
                              void* d_out, int out_size, void* d_ws, size_t ws_size,
                              hipStream_t stream) {
  (void)in_sizes; (void)n_in; (void)out_size; (void)ws_size;
  float* wsf = (float*)d_ws;
  float* out = (float*)d_out;
  const float* X = (const float*)d_in[0];
  const float *bw[8], *bg[8], *bb[8], *hw1[8], *hg1[8], *hb1[8],
              *hw2[8], *hg2[8], *hb2[8], *hw3[8], *hb3[8];
  for (int t = 0; t < 8; ++t) {
    bw[t]  = (const float*)d_in[1 + t];  bg[t]  = (const float*)d_in[9 + t];
    bb[t]  = (const float*)d_in[17 + t];
    hw1[t] = (const float*)d_in[25 + t]; hg1[t] = (const float*)d_in[33 + t];
    hb1[t] = (const float*)d_in[41 + t];
    hw2[t] = (const float*)d_in[49 + t]; hg2[t] = (const float*)d_in[57 + t];
    hb2[t] = (const float*)d_in[65 + t];
    hw3[t] = (const float*)d_in[73 + t]; hb3[t] = (const float*)d_in[81 + t];
  }

  // 1. gray
  k_gray<<<512, 256, 0, stream>>>(X, wsf + GRAY_O);

  // 2. area-pool pyramid (2x2 halving chain)
  k_pool_half<<<cdiv(6 * 128 * 128, 256), 256, 0, stream>>>(X, wsf + POOL_O + poolOffL(7), 128);
  for (int L = 6; L >= 0; --L) {
    int S = 1 << L;
    k_pool_half<<<cdiv(6LL * S * S, 256), 256, 0, stream>>>(
        wsf + POOL_O + poolOffL(L + 1), wsf + POOL_O + poolOffL(L), S);
  }

  // 3. feats[i] = concat(pool, bn_silu(conv(gray)))
  for (int i = 0; i < 8; ++i) {
    int H = 1 << i;
    float* feat = wsf + FEATS_O + H_FOFF[i];
    k_copy_rc<<<cdiv(6LL * H * H, 256), 256, 0, stream>>>(
        wsf + POOL_O + poolOffL(i), feat, H, FCH[i]);
    if (i <= 3) {  // split-K (huge kernels, few outputs)
      long long tsp = 2LL * OUTSC[i] * H * H * KSV[i];
      k_backbone_split<<<cdiv(tsp, 256), 256, 0, stream>>>(
          wsf + GRAY_O, bw[i], wsf + BBACC_O, OUTSC[i], KSV[i], STV[i], H);
      long long tf = 2LL * OUTSC[i] * H * H;
      k_backbone_fin<<<cdiv(tf, 256), 256, 0, stream>>>(
          wsf + BBACC_O, bg[i], bb[i], feat, OUTSC[i], KSV[i], H, FCH[i]);
    } else {
      long long tot = 2LL * OUTSC[i] * H * H;
      k_backbone<<<cdiv(tot, 256), 256, 0, stream>>>(
          wsf + GRAY_O, bw[i], bg[i], bb[i], feat, OUTSC[i], KSV[i], STV[i], PDV[i], H, FCH[i]);
    }
  }

  // 4. heads + normalized residual accumulation
  float* y1 = wsf + Y1_O;
  float* y2 = wsf + Y2_O;
  float* allf = wsf + ALLF_O;
  float* finalBuf = wsf + FINAL_O;
  double* st = (double*)(wsf + STATS_O);
  double* part = (double*)(wsf + PART_O);
  float* prev = wsf + PREVA_O;
  float* spare = wsf + PREVB_O;

  for (int i = 0; i < 8; ++i) {
    int H = 1 << i, P = H * H;
    int K1 = ACCUMC[i], OC1 = K1 / 2, OC2 = OC1 / 2;
    float* Ldst = (i == 0) ? prev : finalBuf;
    for (int b = 0; b < 2; ++b) {
      for (int p0 = 0; p0 < P; p0 += CHUNK) {
        int CP = (P - p0 < CHUNK) ? (P - p0) : CHUNK;
        k_gather<<<cdiv((long long)K1 * CP, 256), 256, 0, stream>>>(
            wsf + FEATS_O, allf, K1, i, b, p0, CP);
        dim3 g1(cdiv(OC1, 32), cdiv(CP, 32));
        k_gemm<<<g1, 32, 0, stream>>>(hw1[i], allf, y1, hg1[i], hb1[i],
                                      OC1, K1, CP, CP, CP, 0);
        dim3 g2(cdiv(OC2, 32), cdiv(CP, 32));
        k_gemm<<<g2, 32, 0, stream>>>(hw2[i], y1, y2, hg2[i], hb2[i],
                                      OC2, OC1, CP, CP, CP, 0);
        dim3 g3(cdiv(1000, 32), cdiv(CP, 32));
        k_gemm<<<g3, 32, 0, stream>>>(hw3[i], y2, Ldst + (long long)b * 1000 * P + p0,
                                      (const float*)nullptr, hb3[i],
                                      1000, OC2, CP, CP, P, 1);
      }
    }
    if (i >= 1) {
      int Hp = H >> 1;
      long long Pp = (long long)Hp * Hp;
      for (int b = 0; b < 2; ++b) {
        k_stats<<<256, 256, 0, stream>>>(prev + (long long)b * 1000 * Pp, 1000LL * Pp, 4.0, part);
        k_stats_fin<<<1, 256, 0, stream>>>(part, 256, st + b * 4);
        k_stats<<<256, 256, 0, stream>>>(finalBuf + (long long)b * 1000 * P, 1000LL * P, 1.0, part);
        k_stats_fin<<<1, 256, 0, stream>>>(part, 256, st + b * 4 + 2);
      }
      float* cur = (i < 7) ? spare : finalBuf;   // i==7 combines in place with li
      long long tot = 2LL * 1000 * P;
      k_combine<<<cdiv(tot, 256), 256, 0, stream>>>(prev, finalBuf, cur, st, H,
                                                    1000.0 * (double)P);
      if (i < 7) { float* t = prev; prev = cur; spare = t; }
      else prev = cur;
    }
  }

  // 5. per-image selection
  int P = 16384;
  float* pmax = wsf + PMAX_O;
  int* argc = (int*)(wsf + ARGC_O);
  float* scores = wsf + SCORES_O;
  unsigned* keys = (unsigned*)(wsf + KEYS_O);
  int* miscI = (int*)(wsf + MISC_O);
  float* miscF = (float*)(wsf + MISC_O + 8);
  for (int b = 0; b < 2; ++b) {
    const float* lb = finalBuf + (long long)b * 1000 * P;
    k_pixmax<<<64, 256, 0, stream>>>(lb, pmax, argc, P);
    k_findpos<<<1, 1024, 0, stream>>>(pmax, argc, miscI, P);
    k_scores<<<64, 256, 0, stream>>>(lb, argc, miscI, scores, keys, P);
    k_median<<<1, 1024, 0, stream>>>(keys, argc, miscI, miscF, P);
    k_argmin<<<1, 1024, 0, stream>>>(scores, argc, miscI, miscF, P);
    k_writeout<<<4, 256, 0, stream>>>(lb, miscI, out + b * 1000, P);
  }
}